// RenderLayer_60155311947793
// MI455X (gfx1250) — compile-verified
//
#include <hip/hip_runtime.h>
#include <hip/hip_bf16.h>

// Problem constants (match reference)
#define B_     32
#define V_     65536
#define F_     131072
// Vertices per wave iteration = 32; groups per wave = 8 -> 256 verts/wave
#define GROUPS_PER_WAVE 8

typedef __attribute__((ext_vector_type(2))) float v2f;
typedef __attribute__((ext_vector_type(8))) float v8f;

// ---------------------------------------------------------------------------
// Pass 1: per-vertex projection using V_WMMA_F32_16X16X4_F32.
//   D(16x16) = A(16x4) x B(4x16) + C
//   A rows 0..2  = [R | -R*campos]   (first WMMA, vertices vbase+0..15)
//   A rows 8..10 = [R | -R*campos]   (second WMMA, vertices vbase+16..31,
//                                     accumulated into same D)
//   B columns    = homogeneous vertices (x,y,z,1)
// After the two accumulated WMMAs, lane L holds camera-space x,y,z of vertex
// vbase+L in d[0],d[1],d[2] (ISA C/D layout: VGPR r -> M=r (lanes 0-15),
// M=8+r (lanes 16-31)). Epilogue (divide/affine/normalize) is pure per-lane.
// ---------------------------------------------------------------------------
__global__ __launch_bounds__(256) void project_kernel(
    const float* __restrict__ mesh,     // (B,V,3)
    const float* __restrict__ focal,    // (B,2)
    const float* __restrict__ princpt,  // (B,2)
    const float* __restrict__ campos,   // (B,3)
    const float* __restrict__ camrot,   // (B,3,3)
    const float* __restrict__ affine,   // (B,2,3)
    float* __restrict__ ws)             // (B,V,3) mesh_2d_norm scratch
{
    const int b    = blockIdx.y;
    const int wave = threadIdx.x >> 5;
    const int lane = threadIdx.x & 31;
    const int r    = lane & 15;          // row index within half-wave
    const bool lo  = lane < 16;

    // ---- per-batch camera setup (uniform across block; SMEM-friendly) ----
    const float* Rb = camrot + b * 9;
    const float c0 = campos[b * 3 + 0];
    const float c1 = campos[b * 3 + 1];
    const float c2 = campos[b * 3 + 2];

    // A1: rows 0..2 hold [R | t],  t = -R*campos
    const int  rr  = (r < 2) ? r : 2;     // clamp for safe loads
    const float Rr0 = Rb[rr * 3 + 0];
    const float Rr1 = Rb[rr * 3 + 1];
    const float Rr2 = Rb[rr * 3 + 2];
    const float tr  = -(Rr0 * c0 + Rr1 * c1 + Rr2 * c2);
    const bool  vA  = (r < 3);
    v2f a1;
    a1.x = vA ? (lo ? Rr0 : Rr2) : 0.0f;  // K=0 (lo) / K=2 (hi)
    a1.y = vA ? (lo ? Rr1 : tr ) : 0.0f;  // K=1 (lo) / K=3 (hi)

    // A2: rows 8..10 hold [R | t]
    const int  r2  = r - 8;
    const int  rr2 = (r2 < 0) ? 0 : ((r2 > 2) ? 2 : r2);
    const float S0 = Rb[rr2 * 3 + 0];
    const float S1 = Rb[rr2 * 3 + 1];
    const float S2 = Rb[rr2 * 3 + 2];
    const float ts = -(S0 * c0 + S1 * c1 + S2 * c2);
    const bool  vB = (r2 >= 0) && (r2 < 3);
    v2f a2;
    a2.x = vB ? (lo ? S0 : S2) : 0.0f;
    a2.y = vB ? (lo ? S1 : ts) : 0.0f;

    // epilogue params
    const float fx  = focal[b * 2 + 0],   fy  = focal[b * 2 + 1];
    const float px  = princpt[b * 2 + 0], py  = princpt[b * 2 + 1];
    const float m00 = affine[b * 6 + 0], m01 = affine[b * 6 + 1], m02 = affine[b * 6 + 2];
    const float m10 = affine[b * 6 + 3], m11 = affine[b * 6 + 4], m12 = affine[b * 6 + 5];

    const int vwaveBase = blockIdx.x * (8 * GROUPS_PER_WAVE * 32)
                        + wave * (GROUPS_PER_WAVE * 32);

    for (int g = 0; g < GROUPS_PER_WAVE; ++g) {
        const int vbase = vwaveBase + g * 32;

        // ---- B matrices: 4x16 homogeneous vertex columns ----
        // lo lanes: VGPR0 = B[0][r] = x(v r), VGPR1 = B[1][r] = y(v r)
        // hi lanes: VGPR0 = B[2][r] = z(v r), VGPR1 = B[3][r] = 1.0
        // mesh is read exactly once -> non-temporal loads keep it out of L2
        // so the ws working set stays resident for pass 2.
        const float* base1 = mesh + ((size_t)b * V_ + (size_t)(vbase + r)) * 3;
        const float* base2 = base1 + 16 * 3;

        v2f bm1, bm2;
        {
            const float* px1 = lo ? (base1 + 0) : (base1 + 2);
            bm1.x = __builtin_nontemporal_load(px1);
            const float y1 = __builtin_nontemporal_load(base1 + 1);
            bm1.y = lo ? y1 : 1.0f;
            const float* px2 = lo ? (base2 + 0) : (base2 + 2);
            bm2.x = __builtin_nontemporal_load(px2);
            const float y2 = __builtin_nontemporal_load(base2 + 1);
            bm2.y = lo ? y2 : 1.0f;
        }

        // ---- two accumulated WMMAs: camera-space coords for 32 vertices ----
        v8f d = {};
        d = __builtin_amdgcn_wmma_f32_16x16x4_f32(
                /*neg_a=*/false, a1, /*neg_b=*/false, bm1,
                /*c_mod=*/(short)0, d, /*reuse_a=*/false, /*reuse_b=*/false);
        d = __builtin_amdgcn_wmma_f32_16x16x4_f32(
                false, a2, false, bm2, (short)0, d, false, false);

        float X = d[0];
        float Y = d[1];
        float Z = d[2];

        // ---- perspective divide + affine + normalize (per-lane VALU) ----
        Z += (Z == 0.0f) ? 1.0e-4f : 0.0f;
        const float invz = __builtin_amdgcn_rcpf(Z);   // v_rcp_f32, ~1 ulp
        const float u = X * invz * fx + px;
        const float v = Y * invz * fy + py;
        const float p2x = m00 * u + m01 * v + m02;
        const float p2y = m10 * u + m11 * v + m12;
        const float xn = p2x * (1.0f / 128.0f) - 1.0f;                 // /IMG_W*2-1
        const float yn = (255.0f - p2y) * (1.0f / 128.0f) - 1.0f;      // (H-1-y)/H*2-1

        float* o = ws + ((size_t)b * V_ + (size_t)(vbase + lane)) * 3;
        o[0] = xn;
        o[1] = yn;
        o[2] = Z;
    }
}

// ---------------------------------------------------------------------------
// Pass 2: gather. out[b,f,j,:] = ws[b, vi[f,j], :].
// Per-batch ws slice (786 KB) and vi (1.5 MB) are L2-resident (192 MB L2),
// so this pass is bound by the 151 MB streaming store. Output stores are
// non-temporal so the streaming writes don't evict ws/vi from L2.
// ---------------------------------------------------------------------------
__global__ __launch_bounds__(256) void gather_kernel(
    const float* __restrict__ ws,   // (B,V,3)
    const int*   __restrict__ vi,   // (F,3)
    float*       __restrict__ out,  // (B,F,3,3)
    int total)                      // B*F*3
{
    const int t = blockIdx.x * blockDim.x + threadIdx.x;
    if (t >= total) return;

    const int j = t % 3;
    const int f = (t / 3) % F_;
    const int b = t / (3 * F_);

    const int idx = vi[f * 3 + j];
    const float* src = ws + ((size_t)b * V_ + (size_t)idx) * 3;
    float* dst = out + (size_t)t * 3;
    __builtin_nontemporal_store(src[0], dst + 0);
    __builtin_nontemporal_store(src[1], dst + 1);
    __builtin_nontemporal_store(src[2], dst + 2);
}

extern "C" void kernel_launch(void* const* d_in, const int* in_sizes, int n_in,
                              void* d_out, int out_size, void* d_ws, size_t ws_size,
                              hipStream_t stream) {
    (void)in_sizes; (void)n_in; (void)out_size; (void)ws_size;

    const float* mesh    = (const float*)d_in[0];
    const float* focal   = (const float*)d_in[1];
    const float* princpt = (const float*)d_in[2];
    const float* campos  = (const float*)d_in[3];
    const float* camrot  = (const float*)d_in[4];
    const float* affine  = (const float*)d_in[5];
    const int*   vi      = (const int*)d_in[6];

    float* out = (float*)d_out;
    float* ws  = (float*)d_ws;     // needs B*V*3*4 = 25,165,824 bytes

    // Pass 1: 8 waves/block, 2048 vertices/block
    dim3 grid1(V_ / 2048, B_, 1);
    project_kernel<<<grid1, 256, 0, stream>>>(mesh, focal, princpt, campos,
                                              camrot, affine, ws);

    // Pass 2: one thread per (b, f, j)
    const int total  = B_ * F_ * 3;           // 12,582,912
    const int blocks = (total + 255) / 256;
    gather_kernel<<<blocks, 256, 0, stream>>>(ws, vi, out, total);
}